// GraphConvClassifier_77979426226619
// MI455X (gfx1250) — compile-verified
//
#include <hip/hip_runtime.h>

#define N_NODES   50000
#define N_EDGES   640000
#define N_GRAPHS  500
#define D         128
#define N_CLASSES 10

typedef __attribute__((ext_vector_type(2))) float v2f;
typedef __attribute__((ext_vector_type(8))) float v8f;

// ---------------- utility kernels ----------------

__global__ void zero_f32(float* __restrict__ p, int n) {
  int i = blockIdx.x * blockDim.x + threadIdx.x;
  if (i < n) p[i] = 0.0f;
}

__global__ void degree_kernel(const int* __restrict__ src, const int* __restrict__ dst,
                              float* __restrict__ deg_out, float* __restrict__ deg_in) {
  int e = blockIdx.x * blockDim.x + threadIdx.x;
  if (e < N_EDGES) {
    atomicAdd(&deg_out[src[e]], 1.0f);
    atomicAdd(&deg_in[dst[e]], 1.0f);
  }
}

// in-place deg -> (deg>0 ? rsqrt(deg) : 0)
__global__ void norm_kernel(float* __restrict__ p, int n) {
  int i = blockIdx.x * blockDim.x + threadIdx.x;
  if (i < n) {
    float d = p[i];
    p[i] = (d > 0.0f) ? rsqrtf(d) : 0.0f;
  }
}

// ---------------- edge scatter: agg[dst] += h[src] * norm_src[src] ----------------
// one wave (32 lanes) per edge, float4 per lane (32*4 = 128 features)

__global__ void scatter_kernel(const float* __restrict__ h, const float* __restrict__ norm_src,
                               const int* __restrict__ src, const int* __restrict__ dst,
                               float* __restrict__ agg) {
  int t = blockIdx.x * blockDim.x + threadIdx.x;
  int e = t >> 5;
  int lane = t & 31;
  if (e >= N_EDGES) return;
  int s = src[e];
  int d = dst[e];
  float ns = norm_src[s];
  const float4* hp = (const float4*)(h + (size_t)s * D);
  float4 v = hp[lane];
  float* ap = agg + (size_t)d * D + lane * 4;
  atomicAdd(ap + 0, v.x * ns);
  atomicAdd(ap + 1, v.y * ns);
  atomicAdd(ap + 2, v.z * ns);
  atomicAdd(ap + 3, v.w * ns);
}

// ---------------- fused GEMM: out = relu((agg * norm_dst[:,None]) @ W + b) ----------------
// f32 WMMA 16x16x4; one wave computes a 16(row) x 128(col) strip.
// W staged into LDS in WMMA-ready layout: [ktile(32)][jtile(8)][lane(32)] float2, 64 KB.

__global__ __launch_bounds__(256) void gemm_norm_bias_relu(
    const float* __restrict__ agg, const float* __restrict__ norm_dst,
    const float* __restrict__ W, const float* __restrict__ bias,
    float* __restrict__ out) {
  __shared__ v2f ldsW[32 * 8 * 32];  // 8192 * 8B = 64 KB

  int tid = threadIdx.x;
  // Stage W: entry i = kt*256 + j*32 + lane holds {W[klo][n], W[klo+1][n]}
  // with half = lane>>4, n = j*16 + (lane&15), klo = kt*4 + 2*half.
  for (int i = tid; i < 32 * 8 * 32; i += 256) {
    int kt   = i >> 8;
    int rem  = i & 255;
    int j    = rem >> 5;
    int lane = rem & 31;
    int half = lane >> 4;
    int n    = j * 16 + (lane & 15);
    int klo  = kt * 4 + 2 * half;
    v2f w;
    w.x = W[klo * D + n];
    w.y = W[(klo + 1) * D + n];
    ldsW[i] = w;
  }
  __syncthreads();

  int wave = tid >> 5;
  int lane = tid & 31;
  int half = lane >> 4;
  int l15  = lane & 15;
  int row_base = (blockIdx.x * 8 + wave) * 16;
  if (row_base >= N_NODES) return;  // uniform per wave; EXEC stays all-ones

  int m = row_base + l15;                 // A-matrix row held by this lane
  float nd = norm_dst[m];                 // fused dst-degree normalization
  const float* arow = agg + (size_t)m * D;

  v8f acc[8] = {};
  for (int kt = 0; kt < 32; ++kt) {
    int klo = kt * 4 + 2 * half;
    v2f a;
    a.x = arow[klo] * nd;
    a.y = arow[klo + 1] * nd;
    const v2f* wrow = &ldsW[(kt * 8) * 32 + lane];
#pragma unroll
    for (int j = 0; j < 8; ++j) {
      v2f b = wrow[j * 32];
      acc[j] = __builtin_amdgcn_wmma_f32_16x16x4_f32(
          /*neg_a=*/false, a, /*neg_b=*/false, b,
          /*c_mod=*/(short)0, acc[j], /*reuse_a=*/false, /*reuse_b=*/false);
    }
  }

  // Epilogue: C/D layout -> VGPR r: lanes0-15 row M=r, lanes16-31 row M=r+8
#pragma unroll
  for (int j = 0; j < 8; ++j) {
    int n = j * 16 + l15;
    float bn = bias[n];
#pragma unroll
    for (int r = 0; r < 8; ++r) {
      int orow = row_base + r + 8 * half;
      float v = acc[j][r] + bn;
      out[(size_t)orow * D + n] = v > 0.0f ? v : 0.0f;
    }
  }
}

// ---------------- graph mean pooling + classifier ----------------

__global__ void pool_kernel(const float* __restrict__ h, const int* __restrict__ gid,
                            float* __restrict__ sums, float* __restrict__ counts) {
  int t = blockIdx.x * blockDim.x + threadIdx.x;
  int node = t >> 5;
  int lane = t & 31;
  if (node >= N_NODES) return;
  int g = gid[node];
  const float4* hp = (const float4*)(h + (size_t)node * D);
  float4 v = hp[lane];
  float* sp = sums + (size_t)g * D + lane * 4;
  atomicAdd(sp + 0, v.x);
  atomicAdd(sp + 1, v.y);
  atomicAdd(sp + 2, v.z);
  atomicAdd(sp + 3, v.w);
  if (lane == 0) atomicAdd(&counts[g], 1.0f);
}

__global__ void hg_kernel(const float* __restrict__ sums, const float* __restrict__ counts,
                          float* __restrict__ hg) {
  int i = blockIdx.x * blockDim.x + threadIdx.x;
  if (i < N_GRAPHS * D) {
    int g = i / D;
    hg[i] = sums[i] / fmaxf(counts[g], 1.0f);
  }
}

__global__ void logits_kernel(const float* __restrict__ hg, const float* __restrict__ Wc,
                              const float* __restrict__ bc, float* __restrict__ logits) {
  int i = blockIdx.x * blockDim.x + threadIdx.x;
  if (i < N_GRAPHS * N_CLASSES) {
    int g = i / N_CLASSES;
    int c = i % N_CLASSES;
    const float* hp = hg + (size_t)g * D;
    float s = bc[c];
#pragma unroll 8
    for (int d = 0; d < D; ++d) s += hp[d] * Wc[d * N_CLASSES + c];
    logits[i] = s;
  }
}

// ---------------- launch ----------------

extern "C" void kernel_launch(void* const* d_in, const int* in_sizes, int n_in,
                              void* d_out, int out_size, void* d_ws, size_t ws_size,
                              hipStream_t stream) {
  const float* x   = (const float*)d_in[0];
  const int*   src = (const int*)d_in[1];
  const int*   dst = (const int*)d_in[2];
  const int*   gid = (const int*)d_in[3];
  const float* Wl[4] = {(const float*)d_in[4], (const float*)d_in[6],
                        (const float*)d_in[8], (const float*)d_in[10]};
  const float* bl[4] = {(const float*)d_in[5], (const float*)d_in[7],
                        (const float*)d_in[9], (const float*)d_in[11]};
  const float* Wc = (const float*)d_in[12];
  const float* bc = (const float*)d_in[13];

  float* ws = (float*)d_ws;
  float* bufA     = ws;                                   // 6.4M floats
  float* bufB     = bufA + (size_t)N_NODES * D;           // 6.4M floats
  float* norms    = bufB + (size_t)N_NODES * D;           // 2 * 50000
  float* norm_src = norms;
  float* norm_dst = norms + N_NODES;
  float* sums     = norm_dst + N_NODES;                   // 500*128
  float* counts   = sums + (size_t)N_GRAPHS * D;          // 500

  float* logits = (float*)d_out;                          // 5000
  float* hg     = logits + N_GRAPHS * N_CLASSES;          // 64000

  const int B = 256;
  auto nb = [](long n, int b) { return (int)((n + b - 1) / b); };

  // degree norms
  zero_f32<<<nb(2 * N_NODES, B), B, 0, stream>>>(norms, 2 * N_NODES);
  degree_kernel<<<nb(N_EDGES, B), B, 0, stream>>>(src, dst, norm_src, norm_dst);
  norm_kernel<<<nb(2 * N_NODES, B), B, 0, stream>>>(norms, 2 * N_NODES);

  // 4 graph-conv layers
  const float* hin = x;
  for (int l = 0; l < 4; ++l) {
    zero_f32<<<nb((long)N_NODES * D, B), B, 0, stream>>>(bufB, N_NODES * D);
    scatter_kernel<<<nb((long)N_EDGES * 32, B), B, 0, stream>>>(hin, norm_src, src, dst, bufB);
    gemm_norm_bias_relu<<<(N_NODES / 16 + 7) / 8, 256, 0, stream>>>(bufB, norm_dst, Wl[l], bl[l], bufA);
    hin = bufA;
  }

  // mean-pool + classifier
  zero_f32<<<nb(N_GRAPHS * D + N_GRAPHS, B), B, 0, stream>>>(sums, N_GRAPHS * D + N_GRAPHS);
  pool_kernel<<<nb((long)N_NODES * 32, B), B, 0, stream>>>(bufA, gid, sums, counts);
  hg_kernel<<<nb(N_GRAPHS * D, B), B, 0, stream>>>(sums, counts, hg);
  logits_kernel<<<nb(N_GRAPHS * N_CLASSES, B), B, 0, stream>>>(hg, Wc, bc, logits);
}